// KnowledgeRouter_15908558864479
// MI455X (gfx1250) — compile-verified
//
#include <hip/hip_runtime.h>
#include <math.h>

typedef __attribute__((ext_vector_type(2))) float v2f;
typedef __attribute__((ext_vector_type(8))) float v8f;

#define S_DIM 256
#define C_DIM 32
#define E_DIM 32
#define N_TOT 512
#define KC 32
#define NCHUNK (S_DIM / KC)

__device__ __forceinline__ float sigmoidf_(float x) { return 1.0f / (1.0f + expf(-x)); }

// ---------------------------------------------------------------------------
// Kernel A: softmax normalizers over the full [S,S] mask (re & im separately),
// then DFT of mask row 0 (-> Mrow) and column 0 (-> Mcol), plus sigmoid tokens.
// ws layout (floats): [0]MrowRe[256] [256]MrowIm [512]McolRe [768]McolIm
//                     [1024]tokARe[32] [1056]tokAIm [1088]tokBRe [1120]tokBIm
// ---------------------------------------------------------------------------
__global__ __launch_bounds__(256) void router_setup(const float* __restrict__ ct,
                                                    const float* __restrict__ cm,
                                                    float* __restrict__ ws) {
  __shared__ float r1[256], r2[256];
  __shared__ float bc[4];
  __shared__ float rowre[256], rowim[256], colre[256], colim[256];
  const int tid = threadIdx.x;

  float mre = -1e30f, mim = -1e30f;
  for (int i = tid; i < S_DIM * S_DIM; i += 256) {
    mre = fmaxf(mre, cm[2 * i]);
    mim = fmaxf(mim, cm[2 * i + 1]);
  }
  r1[tid] = mre; r2[tid] = mim;
  __syncthreads();
  for (int off = 128; off > 0; off >>= 1) {
    if (tid < off) { r1[tid] = fmaxf(r1[tid], r1[tid + off]); r2[tid] = fmaxf(r2[tid], r2[tid + off]); }
    __syncthreads();
  }
  if (tid == 0) { bc[0] = r1[0]; bc[1] = r2[0]; }
  __syncthreads();
  mre = bc[0]; mim = bc[1];

  float sre = 0.0f, sim = 0.0f;
  for (int i = tid; i < S_DIM * S_DIM; i += 256) {
    sre += expf(cm[2 * i] - mre);
    sim += expf(cm[2 * i + 1] - mim);
  }
  r1[tid] = sre; r2[tid] = sim;
  __syncthreads();
  for (int off = 128; off > 0; off >>= 1) {
    if (tid < off) { r1[tid] += r1[tid + off]; r2[tid] += r2[tid + off]; }
    __syncthreads();
  }
  if (tid == 0) { bc[2] = r1[0]; bc[3] = r2[0]; }
  __syncthreads();
  sre = bc[2]; sim = bc[3];

  rowre[tid] = expf(cm[2 * tid] - mre) / sre;
  rowim[tid] = expf(cm[2 * tid + 1] - mim) / sim;
  colre[tid] = expf(cm[2 * (tid * S_DIM)] - mre) / sre;
  colim[tid] = expf(cm[2 * (tid * S_DIM) + 1] - mim) / sim;
  __syncthreads();

  float mrr = 0.0f, mri = 0.0f, mcr = 0.0f, mci = 0.0f;
  const float w0 = -6.283185307179586f / 256.0f;
  for (int t = 0; t < 256; ++t) {
    float ang = w0 * (float)((tid * t) & 255);
    float sn, cs;
    sincosf(ang, &sn, &cs);
    mrr += rowre[t] * cs - rowim[t] * sn;
    mri += rowre[t] * sn + rowim[t] * cs;
    mcr += colre[t] * cs - colim[t] * sn;
    mci += colre[t] * sn + colim[t] * cs;
  }
  ws[tid] = mrr; ws[256 + tid] = mri;
  ws[512 + tid] = mcr; ws[768 + tid] = mci;

  if (tid < E_DIM) {
    const int e = tid;
    ws[1024 + e] = sigmoidf_(ct[((e * 2 + 0) * S_DIM + 0) * 2 + 0]);
    ws[1056 + e] = sigmoidf_(ct[((e * 2 + 0) * S_DIM + 0) * 2 + 1]);
    ws[1088 + e] = sigmoidf_(ct[((e * 2 + 1) * S_DIM + 0) * 2 + 0]);
    ws[1120 + e] = sigmoidf_(ct[((e * 2 + 1) * S_DIM + 0) * 2 + 1]);
  }
}

// ---------------------------------------------------------------------------
// Kernel B: per-sample correlations + top-3 expert selection.
// ---------------------------------------------------------------------------
__global__ __launch_bounds__(256) void router_select(const float* __restrict__ a,
                                                     const float* __restrict__ b,
                                                     const float* __restrict__ ws,
                                                     int* __restrict__ topidx) {
  __shared__ float part[256][6];
  __shared__ float cred[32][4];
  __shared__ float A1[4];
  __shared__ float mag[32];
  const int n = blockIdx.x;
  const int tid = threadIdx.x;
  const int c = tid >> 3, u = tid & 7;
  const float* ap = a + ((size_t)n * C_DIM + c) * S_DIM;
  const float* bp = b + ((size_t)n * C_DIM + c) * S_DIM;

  float sa = 0, sb = 0, par = 0, pai = 0, pbr = 0, pbi = 0;
  for (int i = 0; i < 32; ++i) {
    int s = u * 32 + i;
    float av = ap[s], bv = bp[s];
    sa += av; sb += bv;
    par += av * ws[512 + s]; pai += av * ws[768 + s];  // a . Mcol
    pbr += bv * ws[s];       pbi += bv * ws[256 + s];  // b . Mrow
  }
  part[tid][0] = sa; part[tid][1] = sb;
  part[tid][2] = par; part[tid][3] = pai;
  part[tid][4] = pbr; part[tid][5] = pbi;
  __syncthreads();

  if (tid < 32) {
    float s0 = 0, s1 = 0, s2 = 0, s3 = 0, s4 = 0, s5 = 0;
    for (int j = 0; j < 8; ++j) {
      s0 += part[tid * 8 + j][0]; s1 += part[tid * 8 + j][1];
      s2 += part[tid * 8 + j][2]; s3 += part[tid * 8 + j][3];
      s4 += part[tid * 8 + j][4]; s5 += part[tid * 8 + j][5];
    }
    cred[tid][0] = s0 * s4;
    cred[tid][1] = s0 * s5;
    cred[tid][2] = s1 * s2;
    cred[tid][3] = s1 * s3;
  }
  __syncthreads();
  if (tid == 0) {
    float ar = 0, ai = 0, br = 0, bi = 0;
    for (int cc = 0; cc < 32; ++cc) { ar += cred[cc][0]; ai += cred[cc][1]; br += cred[cc][2]; bi += cred[cc][3]; }
    A1[0] = ar; A1[1] = ai; A1[2] = br; A1[3] = bi;
  }
  __syncthreads();
  if (tid < E_DIM) {
    const int e = tid;
    float tar = ws[1024 + e], tai = ws[1056 + e], tbr = ws[1088 + e], tbi = ws[1120 + e];
    float zr = A1[0] * tar + A1[1] * tai + A1[2] * tbr + A1[3] * tbi;
    float zi = A1[1] * tar - A1[0] * tai + A1[3] * tbr - A1[2] * tbi;
    mag[e] = zr * zr + zi * zi;
  }
  __syncthreads();
  if (tid == 0) {
    int s0 = 0, s1 = -1, s2 = -1;
    float b0 = -1.0f, b1 = -1.0f, b2 = -1.0f;
    for (int e = 0; e < 32; ++e) { float m = mag[e]; if (m > b0) { b0 = m; s0 = e; } }
    for (int e = 0; e < 32; ++e) { if (e == s0) continue; float m = mag[e]; if (m > b1) { b1 = m; s1 = e; } }
    for (int e = 0; e < 32; ++e) { if (e == s0 || e == s1) continue; float m = mag[e]; if (m > b2) { b2 = m; s2 = e; } }
    topidx[n * 3 + 0] = s0; topidx[n * 3 + 1] = s1; topidx[n * 3 + 2] = s2;
  }
}

// ---------------------------------------------------------------------------
// Kernel C: per-sample MoE GEMM on the fp32 WMMA path.
// out[n] = (X[n] @ (W_i0 + W_i1 + W_i2)) / 3, X = a+b, [32x256]@[256x256].
// 8 waves/block; wave w owns N-columns [32w, 32w+32); 2x2 tiles of 16x16.
// - Xs stored with a stride-4 swizzle along s (addr = (s+4c)&255): keeps
//   float4 stores 16B aligned and spreads the 16-lanes-same-column A-frag
//   b64 loads across 16 distinct bank pairs (4m mod 64 distinct, m=0..15).
// - Ws double-buffered (2 x 32 rows): chunk i+1 is prefetched into registers
//   (float4, 3-expert sum) before the 32-WMMA burst of chunk i, then drained
//   into the alternate buffer. LDS total = 32 + 64 = 96 KB (< 320 KB/WGP).
// - Output round-trips through Xs so global stores are coalesced b128.
// ---------------------------------------------------------------------------
__global__ __launch_bounds__(256) void moe_wmma_gemm(const float* __restrict__ a,
                                                     const float* __restrict__ b,
                                                     const float* __restrict__ ew,
                                                     const int* __restrict__ topidx,
                                                     float* __restrict__ out) {
  __shared__ __align__(16) float Xs[C_DIM][S_DIM];     // 32 KB
  __shared__ __align__(16) float Ws[2][KC][S_DIM];     // 64 KB
  const int n = blockIdx.x;
  const int tid = threadIdx.x;
  const size_t nb = (size_t)n * C_DIM * S_DIM;

  const int i0 = topidx[n * 3 + 0];
  const int i1 = topidx[n * 3 + 1];
  const int i2 = topidx[n * 3 + 2];
  const float* __restrict__ w0 = ew + (size_t)i0 * (S_DIM * S_DIM);
  const float* __restrict__ w1 = ew + (size_t)i1 * (S_DIM * S_DIM);
  const float* __restrict__ w2 = ew + (size_t)i2 * (S_DIM * S_DIM);

  float4 pre[8];  // prefetch registers for one summed W chunk (32 floats)

  auto loadChunk = [&](int K0) {
#pragma unroll
    for (int r = 0; r < 8; ++r) {
      const int lin = r * 256 + tid;
      const int k = lin >> 6;
      const int s4 = (lin & 63) << 2;
      const size_t g = (size_t)(K0 + k) * S_DIM + s4;
      const float4 x0 = *(const float4*)(w0 + g);
      const float4 x1 = *(const float4*)(w1 + g);
      const float4 x2 = *(const float4*)(w2 + g);
      pre[r].x = (x0.x + x1.x + x2.x) * (1.0f / 3.0f);
      pre[r].y = (x0.y + x1.y + x2.y) * (1.0f / 3.0f);
      pre[r].z = (x0.z + x1.z + x2.z) * (1.0f / 3.0f);
      pre[r].w = (x0.w + x1.w + x2.w) * (1.0f / 3.0f);
    }
  };
  auto storeChunk = [&](int buf) {
#pragma unroll
    for (int r = 0; r < 8; ++r) {
      const int lin = r * 256 + tid;
      const int k = lin >> 6;
      const int s4 = (lin & 63) << 2;
      *(float4*)&Ws[buf][k][s4] = pre[r];
    }
  };

  // X = a + b, float4-vectorized, stride-4 swizzle along s
#pragma unroll
  for (int r = 0; r < 8; ++r) {
    const int lin = r * 256 + tid;
    const int c = lin >> 6;
    const int s4 = (lin & 63) << 2;
    const float4 av = *(const float4*)(a + nb + c * S_DIM + s4);
    const float4 bv = *(const float4*)(b + nb + c * S_DIM + s4);
    float4 v;
    v.x = av.x + bv.x; v.y = av.y + bv.y; v.z = av.z + bv.z; v.w = av.w + bv.w;
    *(float4*)&Xs[c][(s4 + 4 * c) & (S_DIM - 1)] = v;
  }

  loadChunk(0);
  storeChunk(0);

  const int wave = tid >> 5;
  const int lane = tid & 31;
  const int l16 = lane & 15;
  const int lh = lane >> 4;
  const int ncol = wave * 32;

  v8f acc00 = {}, acc01 = {}, acc10 = {}, acc11 = {};

  int cur = 0;
  for (int ci = 0; ci < NCHUNK; ++ci) {
    __syncthreads();  // Ws[cur] visible; Ws[cur^1] consumers (iter ci-1) done
    if (ci + 1 < NCHUNK) loadChunk((ci + 1) * KC);  // global loads overlap WMMA below
    const int k0 = ci * KC;
#pragma unroll
    for (int kk = 0; kk < KC; kk += 4) {
      const int kf = kk + 2 * lh;  // even
      const int m0 = l16, m1 = 16 + l16;
      const int xi0 = (k0 + kf + 4 * m0) & (S_DIM - 1);  // even -> +1 safe
      const int xi1 = (k0 + kf + 4 * m1) & (S_DIM - 1);
      v2f fa0, fa1, fb0, fb1;
      fa0.x = Xs[m0][xi0];     fa0.y = Xs[m0][xi0 + 1];
      fa1.x = Xs[m1][xi1];     fa1.y = Xs[m1][xi1 + 1];
      fb0.x = Ws[cur][kf][ncol + l16];       fb0.y = Ws[cur][kf + 1][ncol + l16];
      fb1.x = Ws[cur][kf][ncol + 16 + l16];  fb1.y = Ws[cur][kf + 1][ncol + 16 + l16];
      acc00 = __builtin_amdgcn_wmma_f32_16x16x4_f32(false, fa0, false, fb0, (short)0, acc00, false, false);
      acc01 = __builtin_amdgcn_wmma_f32_16x16x4_f32(false, fa0, false, fb1, (short)0, acc01, false, false);
      acc10 = __builtin_amdgcn_wmma_f32_16x16x4_f32(false, fa1, false, fb0, (short)0, acc10, false, false);
      acc11 = __builtin_amdgcn_wmma_f32_16x16x4_f32(false, fa1, false, fb1, (short)0, acc11, false, false);
    }
    if (ci + 1 < NCHUNK) storeChunk(cur ^ 1);
    cur ^= 1;
  }

  // Stage result in Xs (done with A-frags), then coalesced b128 stores.
  __syncthreads();
#pragma unroll
  for (int r = 0; r < 8; ++r) {
    const int m = r + 8 * lh;
    Xs[m][ncol + l16] = acc00[r];
    Xs[m][ncol + 16 + l16] = acc01[r];
    Xs[m + 16][ncol + l16] = acc10[r];
    Xs[m + 16][ncol + 16 + l16] = acc11[r];
  }
  __syncthreads();
#pragma unroll
  for (int r = 0; r < 8; ++r) {
    const int lin = r * 256 + tid;
    const int c = lin >> 6;
    const int s4 = (lin & 63) << 2;
    *(float4*)(out + nb + c * S_DIM + s4) = *(const float4*)&Xs[c][s4];
  }
}

extern "C" void kernel_launch(void* const* d_in, const int* in_sizes, int n_in,
                              void* d_out, int out_size, void* d_ws, size_t ws_size,
                              hipStream_t stream) {
  const float* a = (const float*)d_in[0];   // [8,64,32,256]
  const float* b = (const float*)d_in[1];   // [8,64,32,256]
  const float* ct = (const float*)d_in[2];  // [32,2,256,2]
  const float* cm = (const float*)d_in[3];  // [256,256,2]
  const float* ew = (const float*)d_in[4];  // [32,256,256]
  float* out = (float*)d_out;
  float* ws = (float*)d_ws;                 // 1152 floats of router state
  int* topidx = (int*)(ws + 1152);          // 512*3 ints

  router_setup<<<1, 256, 0, stream>>>(ct, cm, ws);
  router_select<<<N_TOT, 256, 0, stream>>>(a, b, ws, topidx);
  moe_wmma_gemm<<<N_TOT, 256, 0, stream>>>(a, b, ew, topidx, out);
}